// DirectionalGAT_8091718386027
// MI455X (gfx1250) — compile-verified
//
#include <hip/hip_runtime.h>
#include <hip/hip_bf16.h>

typedef __attribute__((ext_vector_type(2))) float v2f;
typedef __attribute__((ext_vector_type(8))) float v8f;

#define BVAL 4
#define VVAL 20000
#define DVAL 10
#define FVAL 32
#define OVAL 32
#define NODES_PER_BLOCK 16
#define ROWS (NODES_PER_BLOCK * DVAL)   // 160 rows = 10 WMMA tiles of 16
#define THREADS 320                     // 10 waves (wave32): one tile per wave
#define XSTR 34                         // LDS row stride: even (b64-aligned) + conflict-free

// ---------------------------------------------------------------------------
// Kernel 1: summed[b,v,f] = sum_d inputs[b,v,d,f]   (node-major flat: node=b*V+v)
// ---------------------------------------------------------------------------
__global__ void sum_dirs_kernel(const float* __restrict__ inputs,
                                float* __restrict__ summed, int nNodes) {
  int tid  = blockIdx.x * blockDim.x + threadIdx.x;
  int node = tid >> 5;
  int f    = tid & 31;
  if (node >= nNodes) return;
  const float* p = inputs + (size_t)node * (DVAL * FVAL) + f;
  float s = 0.0f;
#pragma unroll
  for (int d = 0; d < DVAL; ++d) s += p[d * FVAL];
  summed[((size_t)node << 5) + f] = s;
}

// ---------------------------------------------------------------------------
// Kernel 2: fused gather + mask + dense(F->OUT, WMMA fp32) + relu + mask +
//           attention dot + softmax over D + scale + store
// ---------------------------------------------------------------------------
__global__ __launch_bounds__(THREADS)
void gat_fused_kernel(const float* __restrict__ summed,
                      const float* __restrict__ init,
                      const float* __restrict__ mask,
                      const float* __restrict__ Wk,   // [F,OUT] row-major
                      const float* __restrict__ Wb,   // [OUT]
                      const float* __restrict__ aK,   // [OUT]
                      const int*   __restrict__ adj,
                      const int*   __restrict__ mask_index_p,
                      float* __restrict__ out) {
  __shared__ __align__(16) float Xs[ROWS * XSTR];   // masked gathered inputs
  __shared__ __align__(16) float Ts[ROWS * XSTR];   // transformed
  __shared__ float Ws[FVAL * OVAL];
  __shared__ float Wbs[OVAL];
  __shared__ float aKs[OVAL];
  __shared__ float zms[ROWS];     // 1 - mask
  __shared__ float smk[ROWS];     // -1e7 * mask
  __shared__ int   adjr[ROWS];    // neighbor id, -1 if padded
  __shared__ float logit[ROWS];
  __shared__ float coef[ROWS];

  const int tid   = threadIdx.x;
  const int b     = blockIdx.y;
  const int node0 = blockIdx.x * NODES_PER_BLOCK;
  const int mask_index = *mask_index_p;

  // ---- stage W / bias / a ----
  for (int i = tid; i < FVAL * OVAL; i += THREADS) Ws[i] = Wk[i];
  if (tid < OVAL) { Wbs[tid] = Wb[tid]; aKs[tid] = aK[tid]; }

  // ---- per-row mask / adjacency ----
  if (tid < ROWS) {
    int node = node0 + tid / DVAL;
    int cn   = node < VVAL ? node : VVAL - 1;   // clamp (V%16==0 so unused)
    int d    = tid % DVAL;
    size_t ridx = ((size_t)b * VVAL + cn) * DVAL + d;
    float m  = mask[ridx];
    zms[tid] = 1.0f - m;
    smk[tid] = -1.0e7f * m;
    int av   = adj[ridx];
    adjr[tid] = (av == mask_index || av < 0 || av >= VVAL) ? -1 : av;
  }
  __syncthreads();

  // ---- phase 1: X[r][c] = (1-m) * (gathered_sum + initial_state) ----
  const size_t baseRow = ((size_t)b * VVAL + node0) * DVAL;  // global row index
  for (int i = tid; i < ROWS * FVAL; i += THREADS) {
    int r = i >> 5, c = i & 31;
    int av  = adjr[r];
    float g = (av >= 0) ? summed[(((size_t)b * VVAL + av) << 5) + c] : 0.0f;
    float x = zms[r] * (g + init[((baseRow + (size_t)r) << 5) + c]);
    Xs[r * XSTR + c] = x;
  }
  __syncthreads();

  // ---- phase 2: 16x32 GEMM tile per wave via V_WMMA_F32_16X16X4_F32 ----
  const int wave  = tid >> 5;        // 0..9 -> tile id
  const int lane  = tid & 31;
  const int tileB = wave * 16;
  const int mrow  = lane & 15;       // A: M / B,D: N
  const int khalf = lane >> 4;       // A/B: selects K pair {0,1} vs {2,3}

  float bias0 = Wbs[mrow];           // accumulator vgprs share N -> uniform bias
  float bias1 = Wbs[16 + mrow];
  v8f acc0, acc1;
#pragma unroll
  for (int e = 0; e < 8; ++e) { acc0[e] = bias0; acc1[e] = bias1; }

#pragma unroll
  for (int k0 = 0; k0 < FVAL; k0 += 4) {
    int ka = k0 + 2 * khalf;                      // even -> 8B aligned in LDS
    v2f aF = *(const v2f*)&Xs[(tileB + mrow) * XSTR + ka];
    v2f bF0, bF1;
    bF0.x = Ws[ka * OVAL + mrow];       bF0.y = Ws[(ka + 1) * OVAL + mrow];
    bF1.x = Ws[ka * OVAL + 16 + mrow];  bF1.y = Ws[(ka + 1) * OVAL + 16 + mrow];
    acc0 = __builtin_amdgcn_wmma_f32_16x16x4_f32(false, aF, false, bF0,
                                                 (short)0, acc0, false, false);
    acc1 = __builtin_amdgcn_wmma_f32_16x16x4_f32(false, aF, false, bF1,
                                                 (short)0, acc1, false, false);
  }

  // relu + second zero-mask, stage T in LDS.  D layout: m = e + 8*khalf, n = mrow
#pragma unroll
  for (int e = 0; e < 8; ++e) {
    int r = tileB + e + 8 * khalf;
    float zm = zms[r];
    float t0 = acc0[e]; t0 = (t0 > 0.0f ? t0 : 0.0f) * zm;
    float t1 = acc1[e]; t1 = (t1 > 0.0f ? t1 : 0.0f) * zm;
    Ts[r * XSTR + mrow]      = t0;
    Ts[r * XSTR + 16 + mrow] = t1;
  }
  __syncthreads();

  // ---- phase 3: attention logits + softmax over D per node ----
  if (tid < ROWS) {
    float s = 0.0f;
#pragma unroll
    for (int o = 0; o < OVAL; ++o) s += Ts[tid * XSTR + o] * aKs[o];
    logit[tid] = s + smk[tid];
  }
  __syncthreads();
  if (tid < NODES_PER_BLOCK) {
    float mx = -3.402823466e38f;
#pragma unroll
    for (int d = 0; d < DVAL; ++d) {
      float l = logit[tid * DVAL + d];
      mx = l > mx ? l : mx;
    }
    float ex[DVAL];
    float sum = 0.0f;
#pragma unroll
    for (int d = 0; d < DVAL; ++d) {
      float e = expf(logit[tid * DVAL + d] - mx);
      ex[d] = e; sum += e;
    }
    float inv = 1.0f / sum;
#pragma unroll
    for (int d = 0; d < DVAL; ++d) coef[tid * DVAL + d] = ex[d] * inv;
  }
  __syncthreads();

  // ---- phase 4: out = coef * transformed (coalesced) ----
  const size_t outBase = baseRow << 5;
  for (int i = tid; i < ROWS * FVAL; i += THREADS) {
    int r = i >> 5, c = i & 31;
    out[outBase + (size_t)i] = coef[r] * Ts[r * XSTR + c];
  }
}

// ---------------------------------------------------------------------------
extern "C" void kernel_launch(void* const* d_in, const int* in_sizes, int n_in,
                              void* d_out, int out_size, void* d_ws, size_t ws_size,
                              hipStream_t stream) {
  const float* inputs = (const float*)d_in[0];
  const float* init   = (const float*)d_in[1];
  const float* mask   = (const float*)d_in[2];
  const float* Wk     = (const float*)d_in[3];
  const float* Wb     = (const float*)d_in[4];
  const float* aK     = (const float*)d_in[5];
  const int*   adj    = (const int*)d_in[6];
  const int*   midx   = (const int*)d_in[7];
  float*       out    = (float*)d_out;
  float*       summed = (float*)d_ws;     // B*V*F floats = 10.24 MB scratch

  const int nNodes = BVAL * VVAL;
  dim3 gridA((nNodes * FVAL + 255) / 256);
  sum_dirs_kernel<<<gridA, 256, 0, stream>>>(inputs, summed, nNodes);

  dim3 gridB((VVAL + NODES_PER_BLOCK - 1) / NODES_PER_BLOCK, BVAL);
  gat_fused_kernel<<<gridB, THREADS, 0, stream>>>(summed, init, mask, Wk, Wb,
                                                  aK, adj, midx, out);
}